// FBP_Layer_55808805044890
// MI455X (gfx1250) — compile-verified
//
#include <hip/hip_runtime.h>
#include <stdint.h>

// Geometry constants (match reference)
#define ROWS  512
#define COLS  512
#define NPROJ 720
#define NDET  736
#define NBATCH 32
#define NB    16       // batches per block (2 batch groups)
#define TILE  16       // 16x16 pixel tile per 256-thread block

#define TWO_PI 6.283185307179586f

// ---- CDNA5 async global->LDS copy (ASYNCcnt path) ----------------------
__device__ __forceinline__ void async_b128(uint32_t lds_off, uint32_t voff,
                                           uint64_t sbase) {
  // GVS mode: mem = SGPR64 + VGPR32 ; LDS dest offset in first VGPR
  asm volatile("global_load_async_to_lds_b128 %0, %1, %2"
               :: "v"(lds_off), "v"(voff), "s"(sbase)
               : "memory");
}
__device__ __forceinline__ void wait_async0() {
  asm volatile("s_wait_asynccnt 0" ::: "memory");
}

#define SLAB_FLOATS (NB * NDET)               // 11776 floats / angle slab
#define CHUNKS ((NB * NDET * 4) / 16)         // 2944 x 16B per angle slab
#define COPY_ITERS 12                         // ceil(2944/256)
#define SMEM_BYTES (2 * SLAB_FLOATS * 4 + NPROJ * 8)   // 94208 + 5760 = 99968

__launch_bounds__(256, 2)
__global__ void fbp_bp_kernel(const float* __restrict__ sino,
                              float* __restrict__ out) {
  extern __shared__ float smem[];
  float*  view0 = smem;                        // [NB*NDET]
  float*  view1 = smem + SLAB_FLOATS;          // [NB*NDET]
  float2* trig  = (float2*)(smem + 2 * SLAB_FLOATS);  // [NPROJ]

  const int tid = threadIdx.x;
  const int tx  = tid & (TILE - 1);
  const int ty  = tid >> 4;
  const int col = blockIdx.x * TILE + tx;
  const int row = blockIdx.y * TILE + ty;
  const int bg  = blockIdx.z;                  // batch group 0..1

  // ---- fill trig table (covered by the first barrier in the main loop) ----
  for (int a = tid; a < NPROJ; a += 256) {
    float th = (float)a * (TWO_PI / (float)NPROJ);
    float s, c;
    __sincosf(th, &s, &c);
    trig[a] = make_float2(c, s);
  }

  // ---- angle-invariant per-thread async source offsets -------------------
  // chunk c -> batch b = c/184, 16B offset within detector row = (c%184)*16
  uint32_t voff[COPY_ITERS];
#pragma unroll
  for (int i = 0; i < COPY_ITERS; ++i) {
    uint32_t c = (uint32_t)tid + 256u * (uint32_t)i;
    uint32_t b = c / 184u;
    uint32_t r = c - b * 184u;
    voff[i] = b * (uint32_t)(NPROJ * NDET * 4) + r * 16u;
  }

  const uint64_t sino_base =
      (uint64_t)(uintptr_t)(sino + (size_t)bg * NB * NPROJ * NDET);
  const uint32_t lds0 = (uint32_t)(uintptr_t)view0;
  const uint32_t lds1 = (uint32_t)(uintptr_t)view1;

  // ---- prologue: issue async copy of angle 0 into buffer 0 ---------------
  {
#pragma unroll
    for (int i = 0; i < COPY_ITERS; ++i) {
      uint32_t c = (uint32_t)tid + 256u * (uint32_t)i;
      if (c < CHUNKS) async_b128(lds0 + c * 16u, voff[i], sino_base);
    }
  }

  const float pxf = (float)col - 255.5f;       // (COL-1)/2 centering
  const float pyf = (float)row - 255.5f;

  float acc[NB];
#pragma unroll
  for (int b = 0; b < NB; ++b) acc[b] = 0.0f;

  for (int a = 0; a < NPROJ; ++a) {
    const int cur = a & 1;

    // wait for this wave's slab-a copy, then sync all waves' copies
    wait_async0();
    __syncthreads();

    // issue copy of angle a+1 into the other buffer (overlaps compute below)
    if (a + 1 < NPROJ) {
      const uint64_t sb = sino_base + (uint64_t)(a + 1) * (NDET * 4);
      const uint32_t dst = (cur ? lds0 : lds1);
#pragma unroll
      for (int i = 0; i < COPY_ITERS; ++i) {
        uint32_t c = (uint32_t)tid + 256u * (uint32_t)i;
        if (c < CHUNKS) async_b128(dst + c * 16u, voff[i], sb);
      }
    }

    // ---- geometry (shared by all NB batches) ----
    const float2 cs = trig[a];
    const float c = cs.x, s = cs.y;
    // dot_d = SID - (px*c + py*s);  dot_u = py*c - px*s  (SID terms cancel)
    const float dot_d = 1000.0f - fmaf(pxf, c, pyf * s);
    const float dot_u = fmaf(pyf, c, -pxf * s);
    const float inv_d = __builtin_amdgcn_rcpf(dot_d);     // dot_d in [638,1362] > 0
    const float ui    = fmaf(1500.0f * dot_u, inv_d, 367.5f); // det index (DET_SP=1)
    const float fi    = fminf(fmaxf(floorf(ui), 0.0f), 734.0f);
    const int   i0    = (int)fi;
    const float frac  = fminf(fmaxf(ui - fi, 0.0f), 1.0f);
    const float siv   = 1000.0f * inv_d;
    const float w     = (ui >= 0.0f && ui <= 735.0f) ? siv * siv : 0.0f;
    const float wf    = w * (1.0f - frac);
    const float wc    = w * frac;

    // ---- 2-tap gather from LDS per batch (ds_read2-friendly) ----
    const float* __restrict__ vb = cur ? view1 : view0;
#pragma unroll
    for (int b = 0; b < NB; ++b) {
      const float v0 = vb[b * NDET + i0];
      const float v1 = vb[b * NDET + i0 + 1];
      acc[b] = fmaf(wf, v0, fmaf(wc, v1, acc[b]));
    }
  }

  const float dtheta = TWO_PI / (float)NPROJ;
#pragma unroll
  for (int b = 0; b < NB; ++b) {
    out[((size_t)(bg * NB + b) * ROWS + row) * COLS + col] = acc[b] * dtheta;
  }
}

// ------------------------------------------------------------------------
extern "C" void kernel_launch(void* const* d_in, const int* in_sizes, int n_in,
                              void* d_out, int out_size, void* d_ws, size_t ws_size,
                              hipStream_t stream) {
  (void)in_sizes; (void)n_in; (void)d_ws; (void)ws_size; (void)out_size;
  const float* sino = (const float*)d_in[0];   // [32,1,720,736] f32
  float* out = (float*)d_out;                  // [32,1,512,512] f32

  dim3 block(256, 1, 1);
  dim3 grid(COLS / TILE, ROWS / TILE, NBATCH / NB);  // 32 x 32 x 2
  fbp_bp_kernel<<<grid, block, SMEM_BYTES, stream>>>(sino, out);
}